// LightGNN_52845277610142
// MI455X (gfx1250) — compile-verified
//
#include <hip/hip_runtime.h>
#include <stdint.h>

#define FEAT 64  // D

// ---------------------------------------------------------------------------
// Native device-scope f32 atomic add (no-return form -> STOREcnt, L2 RMW).
// ---------------------------------------------------------------------------
__device__ __forceinline__ void atomic_add_f32_dev(float* p, float v) {
  asm volatile("global_atomic_add_f32 %0, %1, off scope:SCOPE_DEV"
               :: "v"(p), "v"(v));
}

// ---------------------------------------------------------------------------
// Async global -> LDS copy of one dword per lane (CDNA5 ASYNCcnt path).
// ldsOff = low 32 bits of the generic LDS pointer (= LDS byte offset).
// ---------------------------------------------------------------------------
__device__ __forceinline__ void async_g2l_b32(unsigned ldsOff, const int* gptr) {
  asm volatile("global_load_async_to_lds_b32 %0, %1, off"
               :: "v"(ldsOff), "v"(gptr)
               : "memory");
}
__device__ __forceinline__ void wait_async_le(int) {}  // (doc helper, unused)
__device__ __forceinline__ void wait_async0() {
  asm volatile("s_wait_asynccnt 0x0" ::: "memory");
}
__device__ __forceinline__ void wait_async2() {
  asm volatile("s_wait_asynccnt 0x2" ::: "memory");
}

// ---------------------------------------------------------------------------
// Scatter layer: nxt[col] += dinv[row]*dinv[col] * cur[row]
// One wave processes 32-edge chunks. Edge indices are staged into LDS with
// double-buffered async DMA (issue chunk n+stride, process chunk n).
// 16 lanes (half-wave) x float4 cover one 64-float feature row (256 B).
// ---------------------------------------------------------------------------
__global__ void k_scatter(const int* __restrict__ row, const int* __restrict__ col,
                          const float* __restrict__ dinv,
                          const float* __restrict__ cur, float* __restrict__ nxt,
                          int E) {
  __shared__ int   s_idx[8][2][64];   // per wave, per slot: 32 rows + 32 cols
  __shared__ float s_nrm[8][2][32];   // per wave, per slot: 32 edge norms

  const int lane = threadIdx.x & 31;
  const int wid  = threadIdx.x >> 5;

  const int waveId  = (blockIdx.x * blockDim.x + threadIdx.x) >> 5;
  const int nWaves  = (gridDim.x * blockDim.x) >> 5;
  const int nChunks = (E + 31) >> 5;

  const unsigned offR0 = (unsigned)(uintptr_t)&s_idx[wid][0][lane];
  const unsigned offC0 = (unsigned)(uintptr_t)&s_idx[wid][0][32 + lane];
  const unsigned offR1 = (unsigned)(uintptr_t)&s_idx[wid][1][lane];
  const unsigned offC1 = (unsigned)(uintptr_t)&s_idx[wid][1][32 + lane];

  const int half = lane >> 4;          // which edge of the pair this lane works
  const int d0   = (lane & 15) << 2;   // float4 offset within feature row

  int ch   = waveId;
  int slot = 0;
  if (ch < nChunks) {                  // prime slot 0
    int gi = min(ch * 32 + lane, E - 1);
    async_g2l_b32(offR0, row + gi);
    async_g2l_b32(offC0, col + gi);
  }

  for (; ch < nChunks; ch += nWaves, slot ^= 1) {
    const int nxtCh = ch + nWaves;
    if (nxtCh < nChunks) {             // prefetch next chunk into other slot
      int gi = min(nxtCh * 32 + lane, E - 1);
      if (slot == 0) { async_g2l_b32(offR1, row + gi); async_g2l_b32(offC1, col + gi); }
      else           { async_g2l_b32(offR0, row + gi); async_g2l_b32(offC0, col + gi); }
      wait_async2();                   // in-order: current slot's 2 loads done
    } else {
      wait_async0();
    }

    int*   lrow = &s_idx[wid][slot][0];
    int*   lcol = lrow + 32;
    float* lnrm = &s_nrm[wid][slot][0];

    // Each lane computes the norm for one staged edge.
    lnrm[lane] = dinv[lrow[lane]] * dinv[lcol[lane]];

    const int base = ch << 5;
    const int cnt  = min(32, E - base);
    #pragma unroll 4
    for (int s = 0; s < 16; ++s) {
      const int el = (s << 1) + half;  // lanes 0-15 -> even edge, 16-31 -> odd
      if (el < cnt) {
        const int   r  = lrow[el];
        const int   c  = lcol[el];
        const float nm = lnrm[el];
        const float4 v = *(const float4*)(cur + r * FEAT + d0);
        float* dst = nxt + c * FEAT + d0;
        atomic_add_f32_dev(dst + 0, v.x * nm);
        atomic_add_f32_dev(dst + 1, v.y * nm);
        atomic_add_f32_dev(dst + 2, v.z * nm);
        atomic_add_f32_dev(dst + 3, v.w * nm);
      }
    }
  }
}

// ---------------------------------------------------------------------------
// Utility kernels
// ---------------------------------------------------------------------------
__global__ void k_zero(float* __restrict__ p, int n) {
  int i = blockIdx.x * blockDim.x + threadIdx.x;
  if (i < n) p[i] = 0.0f;
}

__global__ void k_deg(const int* __restrict__ col, float* __restrict__ deg, int E) {
  int e = blockIdx.x * blockDim.x + threadIdx.x;
  if (e < E) atomic_add_f32_dev(&deg[col[e]], 1.0f);
}

__global__ void k_rsqrt(float* __restrict__ dinv, int n) {
  int i = blockIdx.x * blockDim.x + threadIdx.x;
  if (i < n) {
    float d = dinv[i];
    dinv[i] = (d > 0.0f) ? rsqrtf(d) : 0.0f;  // d>=1 when >0, so max(d,1)==d
  }
}

// x = emb[n_id], acc = x      (one float4 per thread, 16 threads per node)
__global__ void k_gather(const int* __restrict__ n_id, const float* __restrict__ emb,
                         float* __restrict__ x, float* __restrict__ acc, int N) {
  int t = blockIdx.x * blockDim.x + threadIdx.x;
  if (t >= N * 16) return;
  int node = t >> 4;
  int q = (t & 15) << 2;
  long long src = (long long)n_id[node] * FEAT + q;
  float4 v = *(const float4*)(emb + src);
  int dst = node * FEAT + q;
  *(float4*)(x + dst) = v;
  *(float4*)(acc + dst) = v;
}

// acc += add ; zbuf = 0   (fuses layer accumulation with zeroing the buffer
// the next layer's scatter will atomically accumulate into)
__global__ void k_add_zero(float* __restrict__ acc, const float* __restrict__ add,
                           float* __restrict__ zbuf, int n4) {
  int i = blockIdx.x * blockDim.x + threadIdx.x;
  if (i < n4) {
    float4 a = ((const float4*)acc)[i];
    float4 b = ((const float4*)add)[i];
    a.x += b.x; a.y += b.y; a.z += b.z; a.w += b.w;
    ((float4*)acc)[i] = a;
    float4 z; z.x = 0.f; z.y = 0.f; z.z = 0.f; z.w = 0.f;
    ((float4*)zbuf)[i] = z;
  }
}

// acc = (acc + add) * s     (fused last layer + mean)
__global__ void k_add_scale(float* __restrict__ acc, const float* __restrict__ add,
                            int n4, float s) {
  int i = blockIdx.x * blockDim.x + threadIdx.x;
  if (i < n4) {
    float4 a = ((const float4*)acc)[i];
    float4 b = ((const float4*)add)[i];
    a.x = (a.x + b.x) * s; a.y = (a.y + b.y) * s;
    a.z = (a.z + b.z) * s; a.w = (a.w + b.w) * s;
    ((float4*)acc)[i] = a;
  }
}

// ---------------------------------------------------------------------------
extern "C" void kernel_launch(void* const* d_in, const int* in_sizes, int n_in,
                              void* d_out, int out_size, void* d_ws, size_t ws_size,
                              hipStream_t stream) {
  const int*   n_id = (const int*)d_in[0];
  const int*   edge = (const int*)d_in[1];   // [2, E]: row then col
  const float* emb  = (const float*)d_in[3]; // [1M, 64]

  const int N  = in_sizes[0];
  const int E  = in_sizes[1] / 2;
  const int ND = N * FEAT;
  const int* row = edge;
  const int* col = edge + E;

  // Workspace layout: dinv[N] | bufA[N*64] | bufB[N*64]   (~52 MB, L2 resident)
  float* dinv = (float*)d_ws;
  float* bufA = dinv + (size_t)((N + 255) & ~255);
  float* bufB = bufA + (size_t)ND;
  float* acc  = (float*)d_out;

  const int B = 256;

  k_zero  <<<(N + B - 1) / B, B, 0, stream>>>(dinv, N);
  k_deg   <<<(E + B - 1) / B, B, 0, stream>>>(col, dinv, E);
  k_rsqrt <<<(N + B - 1) / B, B, 0, stream>>>(dinv, N);
  k_gather<<<(N * 16 + B - 1) / B, B, 0, stream>>>(n_id, emb, bufA, acc, N);
  k_zero  <<<(ND + B - 1) / B, B, 0, stream>>>(bufB, ND);

  float* cur = bufA;
  float* nxt = bufB;
  for (int l = 0; l < 3; ++l) {
    k_scatter<<<2048, B, 0, stream>>>(row, col, dinv, cur, nxt, E);
    if (l < 2) {
      // acc += nxt, and zero `cur` (it becomes the scatter target next layer)
      k_add_zero<<<(ND / 4 + B - 1) / B, B, 0, stream>>>(acc, nxt, cur, ND / 4);
    } else {
      k_add_scale<<<(ND / 4 + B - 1) / B, B, 0, stream>>>(acc, nxt, ND / 4, 0.25f);
    }
    float* t = cur; cur = nxt; nxt = t;
  }
}